// FrustumPooling_90520730731183
// MI455X (gfx1250) — compile-verified
//
#include <hip/hip_runtime.h>
#include <hip/hip_bf16.h>

// Problem constants (match reference)
#define BB   2
#define NN   6
#define CC   64
#define HH   28
#define WW   50
#define DD   59
#define BHH  200
#define BWW  200
#define HWP  (HH*WW)            // 1400 pixels per camera
#define NPIX (BB*NN*HH*WW)      // 16800 pixels total
#define CELLS_PER_B (BHH*BWW)   // 40000

typedef __attribute__((ext_vector_type(2))) float v2f;
typedef __attribute__((ext_vector_type(8))) float v8f;

// ---------------------------------------------------------------------------
// Kernel 1: per-camera combined matrices  M = R @ K^-1  and translation t.
// ---------------------------------------------------------------------------
__global__ void frustum_cams_kernel(const float* __restrict__ K,
                                    const float* __restrict__ E,
                                    float* __restrict__ cams) {
    int i = threadIdx.x;
    if (i >= BB*NN) return;
    const float* k = K + i*9;
    float a=k[0], b=k[1], c=k[2], d=k[3], e=k[4], f=k[5], g=k[6], h=k[7], ii=k[8];
    float A0 = e*ii - f*h, A1 = c*h - b*ii, A2 = b*f - c*e;
    float A3 = f*g - d*ii, A4 = a*ii - c*g, A5 = c*d - a*f;
    float A6 = d*h - e*g,  A7 = b*g - a*h,  A8 = a*e - b*d;
    float invdet = 1.0f / (a*A0 + b*A3 + c*A6);
    float inv[9] = {A0*invdet, A1*invdet, A2*invdet,
                    A3*invdet, A4*invdet, A5*invdet,
                    A6*invdet, A7*invdet, A8*invdet};
    const float* ee = E + i*16;
    float R[9] = { ee[0], ee[1], ee[2],
                   ee[4], ee[5], ee[6],
                   ee[8], ee[9], ee[10] };
    float* o = cams + i*12;
    #pragma unroll
    for (int r = 0; r < 3; ++r)
        #pragma unroll
        for (int cc = 0; cc < 3; ++cc)
            o[r*3+cc] = R[r*3+0]*inv[0*3+cc] + R[r*3+1]*inv[1*3+cc] + R[r*3+2]*inv[2*3+cc];
    o[9]  = ee[3];
    o[10] = ee[7];
    o[11] = ee[11];
}

// ---------------------------------------------------------------------------
// Kernel 2: main lift + scatter. One wave32 per pixel (b,n,h,w).
// 59x64 outer product depth (x) feat via V_WMMA_F32_16X16X4_F32 on the matrix
// pipe (co-executes with VALU address math), then per-row f32 atomic scatter.
// FULL=1: acc is (B, BH*BW, C) channel-last workspace  -> contiguous scatter
// FULL=0: acc is d_out directly, channel-major (fallback when ws too small)
// ---------------------------------------------------------------------------
template<int FULL>
__global__ __launch_bounds__(256)
void frustum_scatter_kernel(const float* __restrict__ feat,
                            const float* __restrict__ depth,
                            const float* __restrict__ cams,
                            float* __restrict__ acc) {
    const int lane = threadIdx.x & 31;
    const int wave = threadIdx.x >> 5;
    const int pix  = blockIdx.x * 8 + wave;     // grid sized exactly: < NPIX
    const int w    = pix % WW;
    const int hw   = pix % HWP;
    const int h    = hw / WW;
    const int bn   = pix / HWP;                 // 0..11
    const int b    = bn / NN;

    const float* Mt = cams + bn*12;
    const float fw = (float)w, fh = (float)h;
    const float dirx = Mt[0]*fw + Mt[1]*fh + Mt[2];
    const float diry = Mt[3]*fw + Mt[4]*fh + Mt[5];
    const float dirz = Mt[6]*fw + Mt[7]*fh + Mt[8];
    const float tx = Mt[9], ty = Mt[10], tz = Mt[11];

    const float* featp  = feat  + (size_t)bn * CC * HWP + hw;  // channel stride HWP
    const float* depthp = depth + (size_t)bn * DD * HWP + hw;  // bin stride HWP

    __builtin_prefetch(featp,  0, 1);   // global_prefetch_b8
    __builtin_prefetch(depthp, 0, 1);

    // Shared base of both output layouts: b * (C * BH * BW)
    const int bBase = b * (CC * CELLS_PER_B);
    const int rowAdd = (lane >> 4) << 3;        // lanes 16-31 hold rows M=r+8

    // B operands: feat chunks of 16 channels in row K=0 (VGPR0, lanes 0-15)
    v2f bmat[4];
    #pragma unroll
    for (int ck = 0; ck < 4; ++ck) {
        v2f bm = {0.f, 0.f};
        if (lane < 16) bm.x = featp[(ck*16 + lane) * HWP];
        bmat[ck] = bm;
    }

    #pragma unroll 1                            // keep I$ footprint small
    for (int dk = 0; dk < 4; ++dk) {
        // A operand: depth chunk of 16 bins in column K=0 (VGPR0, lanes 0-15)
        v2f amat = {0.f, 0.f};
        const int dl = dk*16 + lane;
        if (lane < 16 && dl < DD) amat.x = depthp[dl * HWP];

        #pragma unroll
        for (int ck = 0; ck < 4; ++ck) {
            v8f prod = {};
            // D[m,n] = depth[dk*16+m] * feat[ck*16+n]   (exact outer product)
            prod = __builtin_amdgcn_wmma_f32_16x16x4_f32(
                false, amat, false, bmat[ck], (short)0, prod, false, false);

            const int c = ck*16 + (lane & 15);          // column N = lane % 16
            // Hoisted channel term of the output index
            const int cTerm = FULL ? c : c * CELLS_PER_B;

            #pragma unroll
            for (int r = 0; r < 8; ++r) {
                // C layout: VGPR r holds row M=r (lanes 0-15), M=r+8 (lanes 16-31)
                const int myd = dk*16 + r + rowAdd;
                const float db = (float)(myd + 1);      // depth bins are 1..59
                const float px = db*dirx + tx;
                const float py = db*diry + ty;
                const float pz = db*dirz + tz;
                // (x - X0)/RES with truncation toward zero (matches astype(int32))
                const int bx = (int)((px + 50.0f) * 2.0f);
                const int by = (int)((py + 50.0f) * 2.0f);
                if ((unsigned)bx < (unsigned)BWW && (unsigned)by < (unsigned)BHH &&
                    pz > 0.0f && myd < DD) {
                    const int cell = by*BWW + bx;
                    const int idx  = FULL ? (bBase + (cell << 6) + cTerm)
                                          : (bBase + cTerm + cell);
                    __hip_atomic_fetch_add(acc + idx, prod[r],
                                           __ATOMIC_RELAXED,
                                           __HIP_MEMORY_SCOPE_AGENT);
                }
            }
        }
    }
}

// ---------------------------------------------------------------------------
// Kernel 3: LDS-tiled transpose (B, BH*BW, C) -> (B, C, BH*BW).
// 64 cells x 64 channels per block, padded LDS stride 65 -> bank-conflict free.
// ---------------------------------------------------------------------------
__global__ __launch_bounds__(256)
void frustum_transpose_kernel(const float* __restrict__ acc,
                              float* __restrict__ out) {
    __shared__ float tile[64 * 65];
    const int tid      = threadIdx.x;
    const int b        = blockIdx.x / (CELLS_PER_B / 64);        // 625 tiles per b
    const int cellBase = (blockIdx.x % (CELLS_PER_B / 64)) * 64;
    const size_t srcBase = ((size_t)b * CELLS_PER_B + cellBase) * CC;

    #pragma unroll
    for (int i = 0; i < 16; ++i) {
        const int idx = i*256 + tid;                // 4096 contiguous floats
        tile[(idx >> 6)*65 + (idx & 63)] = acc[srcBase + idx];
    }
    __syncthreads();
    #pragma unroll
    for (int i = 0; i < 16; ++i) {
        const int idx  = i*256 + tid;
        const int c    = idx >> 6;
        const int cell = idx & 63;
        out[((size_t)(b*CC + c)) * CELLS_PER_B + cellBase + cell] = tile[cell*65 + c];
    }
}

// ---------------------------------------------------------------------------
// Launch
// inputs: [0] features (B,N,C,H,W) f32, [1] depth (B,N,D,H,W) f32,
//         [2] intrinsics (B,N,3,3) f32, [3] extrinsics (B,N,4,4) f32
// output: (B, C, BH, BW) f32
// ---------------------------------------------------------------------------
extern "C" void kernel_launch(void* const* d_in, const int* in_sizes, int n_in,
                              void* d_out, int out_size, void* d_ws, size_t ws_size,
                              hipStream_t stream) {
    const float* feat  = (const float*)d_in[0];
    const float* depth = (const float*)d_in[1];
    const float* K     = (const float*)d_in[2];
    const float* E     = (const float*)d_in[3];
    float* out = (float*)d_out;
    char*  ws  = (char*)d_ws;

    float* cams = (float*)ws;                       // 12 * 12 floats = 576 B
    const size_t ACC_OFF   = 1024;
    const size_t ACC_BYTES = (size_t)BB * CELLS_PER_B * CC * sizeof(float); // 20.48 MB
    const bool full = (ws_size >= ACC_OFF + ACC_BYTES);

    float* acc = full ? (float*)(ws + ACC_OFF) : out;

    if (full) {
        hipMemsetAsync(acc, 0, ACC_BYTES, stream);          // graph-capture safe
    } else {
        hipMemsetAsync(out, 0, (size_t)out_size * sizeof(float), stream);
    }

    frustum_cams_kernel<<<1, 32, 0, stream>>>(K, E, cams);

    // 16800 pixels, 8 waves (256 threads) per block -> exactly 2100 blocks
    if (full) {
        frustum_scatter_kernel<1><<<NPIX/8, 256, 0, stream>>>(feat, depth, cams, acc);
        frustum_transpose_kernel<<<BB * (CELLS_PER_B/64), 256, 0, stream>>>(acc, out);
    } else {
        frustum_scatter_kernel<0><<<NPIX/8, 256, 0, stream>>>(feat, depth, cams, acc);
    }
}